// Model_17145509445950
// MI455X (gfx1250) — compile-verified
//
#include <hip/hip_runtime.h>
#include <math.h>

// Problem constants (match reference)
#define BB 32
#define PP 131072
#define DD 16
#define OD 8

typedef __attribute__((ext_vector_type(2))) float v2f;
typedef __attribute__((ext_vector_type(8))) float v8f;

// ---------------------------------------------------------------------------
// Kernel 0: build constant matrices, d-major (row = K dim d, col = o),
// cols 8..15 zero-padded.
//   M1[d][o] = sum_e A[e][d] * C[o][e]   (folds pred matmul into obs matmul)
//   M2[d][o] = 0.1 * C[o][d]             (process-noise contribution)
// These feed the WMMA A-operand as M1^T / M2^T (rows o, cols d).
// ---------------------------------------------------------------------------
__global__ void pf_prep(const float* __restrict__ A, const float* __restrict__ C,
                        float* __restrict__ M1, float* __restrict__ M2) {
    int t = threadIdx.x;          // 256 threads -> one (d, n) cell each
    int d = t >> 4;
    int n = t & 15;
    float m1 = 0.0f, m2 = 0.0f;
    if (n < OD) {
        #pragma unroll
        for (int e = 0; e < DD; ++e) m1 += A[e * DD + d] * C[n * DD + e];
        m2 = 0.1f * C[n * DD + d];
    }
    M1[d * 16 + n] = m1;
    M2[d * 16 + n] = m2;
}

// ---------------------------------------------------------------------------
// Kernel 1: likelihood via WMMA f32 16x16x4 chains, transposed mapping:
//   D[o][particle] = M1^T x X^T + M2^T x E^T
// A-operand (16x4 per k-step) = M1^T/M2^T slices: constant per wave.
// B-operand (4x16 per k-step) = particle/eps tile: lane n supplies column
//   (particle p0+n), halves supply K-row pairs -> same float2 loads as a
//   row-major tile read.
// C/D layout: lane n (n<16, half=0), VGPR v -> element (M=o=v, N=n), so all
// 8 valid obs rows of particle p0+n live in ONE lane -> per-lane reduction,
// no cross-lane shuffles, coalesced lw load / nlw store.
// ---------------------------------------------------------------------------
#define LL_CHUNK 1024            // particles per block (8 waves x 8 tiles x 16)
__global__ void pf_loglik(const float* __restrict__ part,
                          const float* __restrict__ eps,
                          const float* __restrict__ lw,
                          const float* __restrict__ obs,
                          const float* __restrict__ M1,
                          const float* __restrict__ M2,
                          float* __restrict__ nlw) {
    int blk   = blockIdx.x;
    int b     = blk >> 7;                     // 128 blocks per batch
    int chunk = blk & 127;
    int tid   = threadIdx.x;
    int wave  = tid >> 5;
    int lane  = tid & 31;
    int n     = lane & 15;
    int half  = lane >> 4;

    const float* pb  = part + (size_t)b * PP * DD;
    const float* eb  = eps  + (size_t)b * PP * DD;
    const float* lwb = lw   + (size_t)b * PP;
    float*       nb  = nlw  + (size_t)b * PP;

    // Constant A operands (M1^T, M2^T): load once, reuse for every tile.
    // A-layout: lanes 0-15 hold M=lane (VGPR0=K0,VGPR1=K1); lanes 16-31 hold
    // M=lane-16 (VGPR0=K2,VGPR1=K3). M = o index, K = d.
    v2f A1[4], A2[4];
    #pragma unroll
    for (int k = 0; k < 4; ++k) {
        int r = 4 * k + 2 * half;
        A1[k].x = M1[(r    ) * 16 + n];
        A1[k].y = M1[(r + 1) * 16 + n];
        A2[k].x = M2[(r    ) * 16 + n];
        A2[k].y = M2[(r + 1) * 16 + n];
    }
    // Observation: wave-uniform scalars.
    float obss[OD];
    #pragma unroll
    for (int v = 0; v < OD; ++v) obss[v] = obs[b * OD + v];

    int p0w = chunk * LL_CHUNK + wave * 128;  // 8 tiles of 16 per wave
    for (int t = 0; t < 8; ++t) {
        int p0 = p0w + t * 16;
        const float* prow = pb + (size_t)(p0 + n) * DD;  // column N = n
        const float* erow = eb + (size_t)(p0 + n) * DD;

        v8f c = {};
        #pragma unroll
        for (int k = 0; k < 4; ++k) {
            v2f bv;
            int o = 4 * k + 2 * half;
            bv.x = prow[o];
            bv.y = prow[o + 1];
            c = __builtin_amdgcn_wmma_f32_16x16x4_f32(
                    false, A1[k], false, bv, (short)0, c, false, false);
        }
        #pragma unroll
        for (int k = 0; k < 4; ++k) {
            v2f bv;
            int o = 4 * k + 2 * half;
            bv.x = erow[o];
            bv.y = erow[o + 1];
            c = __builtin_amdgcn_wmma_f32_16x16x4_f32(
                    false, A2[k], false, bv, (short)0, c, false, false);
        }

        // Per-lane log-likelihood: lane n holds obs_pred[o=v][p0+n] in c[v].
        float ll = 0.0f;
        #pragma unroll
        for (int v = 0; v < OD; ++v) {
            float dv = obss[v] - c[v];
            ll = fmaf(dv, dv, ll);
        }
        if (half == 0) {
            int m = p0 + n;
            nb[m] = lwb[m] - 2.0f * ll;   // -0.5/sigma^2 = -2
        }
    }
}

// ---------------------------------------------------------------------------
// Kernel 2: per-chunk online logsumexp partials (2048 elements per block).
// ---------------------------------------------------------------------------
__global__ void pf_lsep(const float* __restrict__ nlw,
                        float* __restrict__ pm, float* __restrict__ ps) {
    int blk = blockIdx.x;                     // b*64 + cb ; chunks tile nlw
    const float* x = nlw + (size_t)blk * 2048;
    int tid = threadIdx.x;
    float m = -INFINITY, s = 0.0f;
    for (int i = tid; i < 2048; i += 256) {
        float v = x[i];
        if (v > m) { s = s * expf(m - v) + 1.0f; m = v; }
        else       { s += expf(v - m); }
    }
    __shared__ float sm[256], ss[256];
    sm[tid] = m; ss[tid] = s;
    __syncthreads();
    for (int off = 128; off > 0; off >>= 1) {
        if (tid < off) {
            float m2 = sm[tid + off], s2 = ss[tid + off];
            float M = fmaxf(sm[tid], m2);
            ss[tid] = ss[tid] * expf(sm[tid] - M) + s2 * expf(m2 - M);
            sm[tid] = M;
        }
        __syncthreads();
    }
    if (tid == 0) { pm[blk] = sm[0]; ps[blk] = ss[0]; }
}

// ---------------------------------------------------------------------------
// Kernel 3: one block per batch. Merge 64 (m,s) pairs -> logZ; derive chunk
// weight-sums algebraically (exp(m - logZ) * s, no nlw re-read) and produce
// their inclusive scan for the cumsum pass.
// ---------------------------------------------------------------------------
__global__ void pf_combine(const float* __restrict__ pm, const float* __restrict__ ps,
                           float* __restrict__ logZ, float* __restrict__ bsums) {
    int b = blockIdx.x, t = threadIdx.x;      // 64 threads
    float m = pm[b * 64 + t], s = ps[b * 64 + t];
    __shared__ float sm[64], ss[64], sc[64];
    __shared__ float sz;
    sm[t] = m; ss[t] = s;
    __syncthreads();
    for (int off = 32; off > 0; off >>= 1) {
        if (t < off) {
            float m2 = sm[t + off], s2 = ss[t + off];
            float M = fmaxf(sm[t], m2);
            ss[t] = ss[t] * expf(sm[t] - M) + s2 * expf(m2 - M);
            sm[t] = M;
        }
        __syncthreads();
    }
    if (t == 0) { sz = sm[0] + logf(ss[0]); logZ[b] = sz; }
    __syncthreads();
    sc[t] = expf(m - sz) * s;                 // chunk sum of normalized weights
    __syncthreads();
    for (int off = 1; off < 64; off <<= 1) {
        float v = (t >= off) ? sc[t - off] : 0.0f;
        __syncthreads();
        sc[t] += v;
        __syncthreads();
    }
    bsums[b * 64 + t] = sc[t];                // inclusive
}

// ---------------------------------------------------------------------------
// Kernel 4: final cumsum. Block = 2048 elements, thread t owns [t*8, t*8+8).
// ---------------------------------------------------------------------------
__global__ void pf_cscan(const float* __restrict__ nlw, const float* __restrict__ logZ,
                         const float* __restrict__ bsums, float* __restrict__ cum) {
    int blk = blockIdx.x;
    int b = blk >> 6;
    int cb = blk & 63;
    float z = logZ[b];
    const float* x = nlw + (size_t)blk * 2048;
    float*       y = cum + (size_t)blk * 2048;
    int tid = threadIdx.x;
    float w[8], run = 0.0f;
    #pragma unroll
    for (int i = 0; i < 8; ++i) { run += expf(x[tid * 8 + i] - z); w[i] = run; }
    __shared__ float sh[256];
    sh[tid] = run;
    __syncthreads();
    for (int off = 1; off < 256; off <<= 1) {
        float v = (tid >= off) ? sh[tid - off] : 0.0f;
        __syncthreads();
        sh[tid] += v;
        __syncthreads();
    }
    float excl = (tid > 0) ? sh[tid - 1] : 0.0f;
    float boff = (cb > 0) ? bsums[b * 64 + cb - 1] : 0.0f;
    #pragma unroll
    for (int i = 0; i < 8; ++i) y[tid * 8 + i] = w[i] + excl + boff;
}

// ---------------------------------------------------------------------------
// Kernel 5: systematic resampling. Binary search (searchsorted side='left'),
// clip, recompute pred row for the gathered index (A in LDS), write outputs.
// ---------------------------------------------------------------------------
__global__ void pf_resample(const float* __restrict__ cum,
                            const float* __restrict__ part,
                            const float* __restrict__ eps,
                            const float* __restrict__ A,
                            const float* __restrict__ urand,
                            float* __restrict__ out) {
    __shared__ float sA[256];
    int tid = threadIdx.x;
    sA[tid] = A[tid];                 // 16x16 state matrix
    __syncthreads();

    int blk = blockIdx.x;
    int b  = blk >> 9;                // 512 blocks per batch
    int jb = ((blk & 511) << 8) + tid;

    float u = ((float)jb + urand[b]) * (1.0f / (float)PP);
    const float* cb = cum + (size_t)b * PP;
    int lo = 0, hi = PP;
    while (lo < hi) {                 // lower_bound: first i with cum[i] >= u
        int mid = (lo + hi) >> 1;
        if (cb[mid] < u) lo = mid + 1; else hi = mid;
    }
    int idx = lo < (PP - 1) ? lo : (PP - 1);

    const float4* xr = (const float4*)(part + ((size_t)b * PP + idx) * DD);
    const float4* er = (const float4*)(eps  + ((size_t)b * PP + idx) * DD);
    float x[16], e[16];
    #pragma unroll
    for (int q = 0; q < 4; ++q) {
        float4 xv = xr[q], ev = er[q];
        x[4*q+0] = xv.x; x[4*q+1] = xv.y; x[4*q+2] = xv.z; x[4*q+3] = xv.w;
        e[4*q+0] = ev.x; e[4*q+1] = ev.y; e[4*q+2] = ev.z; e[4*q+3] = ev.w;
    }
    float4* orow = (float4*)(out + ((size_t)b * PP + jb) * DD);
    #pragma unroll
    for (int q = 0; q < 4; ++q) {
        float4 r;
        float* rp = (float*)&r;
        #pragma unroll
        for (int j = 0; j < 4; ++j) {
            int ee = 4 * q + j;
            float acc = 0.1f * e[ee];
            #pragma unroll
            for (int d = 0; d < DD; ++d) acc += sA[ee * DD + d] * x[d];
            rp[j] = acc;
        }
        orow[q] = r;
    }
    float* out2 = out + (size_t)BB * PP * DD;
    out2[(size_t)b * PP + jb] = -11.7835020695f;   // -log(131072)
}

// ---------------------------------------------------------------------------
extern "C" void kernel_launch(void* const* d_in, const int* in_sizes, int n_in,
                              void* d_out, int out_size, void* d_ws, size_t ws_size,
                              hipStream_t stream) {
    (void)in_sizes; (void)n_in; (void)out_size; (void)ws_size;
    const float* particles = (const float*)d_in[0];   // (B,P,D)
    const float* log_w     = (const float*)d_in[1];   // (B,P)
    const float* obs       = (const float*)d_in[2];   // (B,OD)
    const float* A         = (const float*)d_in[3];   // (D,D)
    const float* C         = (const float*)d_in[4];   // (OD,D)
    const float* eps       = (const float*)d_in[5];   // (B,P,D)
    const float* urand     = (const float*)d_in[6];   // (B,1)
    float* out = (float*)d_out;

    float* wsf   = (float*)d_ws;
    float* M1    = wsf;                                 // 256
    float* M2    = wsf + 256;                           // 256
    float* nlw   = wsf + 512;                           // B*P
    float* logZ  = nlw + (size_t)BB * PP;               // B
    float* bsums = logZ + BB;                           // B*64
    float* pm    = bsums + (size_t)BB * 64;             // B*64
    float* ps    = pm + (size_t)BB * 64;                // B*64
    float* cum   = ps + (size_t)BB * 64;                // B*P

    pf_prep<<<1, 256, 0, stream>>>(A, C, M1, M2);
    pf_loglik<<<BB * (PP / LL_CHUNK), 256, 0, stream>>>(particles, eps, log_w, obs,
                                                        M1, M2, nlw);
    pf_lsep<<<BB * 64, 256, 0, stream>>>(nlw, pm, ps);
    pf_combine<<<BB, 64, 0, stream>>>(pm, ps, logZ, bsums);
    pf_cscan<<<BB * 64, 256, 0, stream>>>(nlw, logZ, bsums, cum);
    pf_resample<<<BB * (PP / 256), 256, 0, stream>>>(cum, particles, eps, A, urand, out);
}